// ActionNetwork_66133906423959
// MI455X (gfx1250) — compile-verified
//
#include <hip/hip_runtime.h>

#define NN 8
#define FF 88
#define ROWSTRIDE 90   // dwords; bank index 26*t mod 64 injective over a wave -> conflict-free LDS reads
#define BLOCK 128

__global__ __launch_bounds__(BLOCK) void actionnet_kernel(
    const float* __restrict__ x,
    const float* __restrict__ W0,
    const float* __restrict__ b0,
    const float* __restrict__ W1,
    const float* __restrict__ b1,
    const float* __restrict__ dp,
    const float* __restrict__ qp,
    float* __restrict__ out,
    int EP)
{
    __shared__ __align__(16) float lds_x[BLOCK * ROWSTRIDE]; // 46080 B
    __shared__ float par[360];                               // W0|b0|W1|b1|dp|qp

    const int tid = threadIdx.x;
    const int t   = blockIdx.x * BLOCK + tid;
    const bool active = (t < EP);
    const int ts  = active ? t : 0;

    // ---- CDNA5 async global -> LDS staging of this thread's 88-float sample row.
    // One LDS-base VGPR + one global-address VGPR pair; the shared 24-bit
    // INST_OFFSET advances both sides.
    {
        const float* g = x + (size_t)ts * FF;
        unsigned l = (unsigned)(size_t)(&lds_x[tid * ROWSTRIDE]); // low 32 bits == LDS offset
#define AL(OFF) asm volatile("global_load_async_to_lds_b64 %0, %1, off offset:" #OFF \
                             :: "v"(l), "v"(g) : "memory")
        AL(0);   AL(8);   AL(16);  AL(24);  AL(32);  AL(40);  AL(48);  AL(56);
        AL(64);  AL(72);  AL(80);  AL(88);  AL(96);  AL(104); AL(112); AL(120);
        AL(128); AL(136); AL(144); AL(152); AL(160); AL(168); AL(176); AL(184);
        AL(192); AL(200); AL(208); AL(216); AL(224); AL(232); AL(240); AL(248);
        AL(256); AL(264); AL(272); AL(280); AL(288); AL(296); AL(304); AL(312);
        AL(320); AL(328); AL(336); AL(344);
#undef AL
    }

    // ---- parameters into LDS (broadcast reads later) ----
    for (int p = tid; p < 152; p += BLOCK) par[p] = W0[p];
    for (int p = tid; p < 64;  p += BLOCK) {
        par[160 + p] = W1[p];
        par[232 + p] = dp[p];
        par[296 + p] = qp[p];
    }
    if (tid < 8)       par[152 + tid]     = b0[tid];
    else if (tid < 16) par[224 + tid - 8] = b1[tid - 8];

    __syncthreads();
    asm volatile("s_wait_asynccnt 0x0" ::: "memory");

    const float* lx = &lds_x[tid * ROWSTRIDE];

    float veh[NN];
#pragma unroll
    for (int i = 0; i < NN; ++i) veh[i] = lx[i];

    // pot0[n] = b0[n] + W0[n,:] . res[n,:]
    // res per n: [veh[n], x[8+8n+0], x[8+8n+1], queue[n][0..7], queue[0..7][n]]
    float pot0[NN];
#pragma unroll
    for (int n = 0; n < NN; ++n) {
        float s = par[152 + n];
        s += par[n * 19 + 0] * veh[n];
        s += par[n * 19 + 1] * lx[8 + 8 * n + 0];
        s += par[n * 19 + 2] * lx[8 + 8 * n + 1];
#pragma unroll
        for (int j = 0; j < NN; ++j)
            s += par[n * 19 + 3 + j] * lx[24 + n * 8 + j];
#pragma unroll
        for (int j = 0; j < NN; ++j)
            s += par[n * 19 + 11 + j] * lx[24 + j * 8 + n];
        pot0[n] = s;
    }

    float pot[NN];
#pragma unroll
    for (int i = 0; i < NN; ++i) {
        float s = par[224 + i];
#pragma unroll
        for (int j = 0; j < NN; ++j)
            s += par[160 + i * 8 + j] * pot0[j];
        pot[i] = s;
    }

    float val[NN * NN];
    float s_i[NN], fv[NN], arr[NN];

    // arrival init = mini_g.sum(axis=1)  (mini_g[n][m] = x[8 + 2n + m])
#pragma unroll
    for (int j = 0; j < NN; ++j)
        arr[j] = lx[8 + 2 * j] + lx[8 + 2 * j + 1];

    float* o = out + (size_t)t * 128;

#pragma unroll
    for (int i = 0; i < NN; ++i) {
        float rs = 0.f;
#pragma unroll
        for (int j = 0; j < NN; ++j) {
            float v = 0.f;
            if (j != i) {
                float d = fmaxf(pot[i] - pot[j], 0.f);
                v = d * par[232 + i * 8 + j] + lx[24 + i * 8 + j] * par[296 + i * 8 + j];
                rs += v;
            }
            val[i * 8 + j] = v;
        }
        float rr  = fmaxf(veh[i] - rs, 0.f);   // relu(remain)
        float inv = 1.0f / (rs + rr);          // 1/denom[i]

        float act[NN];
#pragma unroll
        for (int j = 0; j < NN; ++j)
            act[j] = ((j == i) ? rr : val[i * 8 + j]) * inv;
        if (active) {
            *(float4*)(o + i * 16)     = make_float4(act[0], act[1], act[2], act[3]);
            *(float4*)(o + i * 16 + 4) = make_float4(act[4], act[5], act[6], act[7]);
        }

        float si = veh[i] * inv;               // raw_action[i,j] = val[i,j]*si
        s_i[i] = si;
        fv[i]  = veh[i] - rs * si;             // veh - departure
#pragma unroll
        for (int j = 0; j < NN; ++j)
            if (j != i) arr[j] += val[i * 8 + j] * si;  // arrival accumulation
    }

#pragma unroll
    for (int i = 0; i < NN; ++i) fv[i] += arr[i];       // future_vehicle

#pragma unroll
    for (int i = 0; i < NN; ++i) {
        float si = s_i[i];
        float fgrow[NN];
        float it = 0.f;
#pragma unroll
        for (int j = 0; j < NN; ++j) {
            float v  = val[i * 8 + j];
            float fg = fmaxf(v - v * si, 0.f);          // future_gradient
            fgrow[j] = fg;
            it += fg;
        }
        float add = fmaxf(fv[i] - it, 0.f) / 7.0f;      // remain2/(N-1)

        float pr[NN];
#pragma unroll
        for (int j = 0; j < NN; ++j) {
            float v     = val[i * 8 + j];
            float fq    = fmaxf(lx[24 + i * 8 + j] - v * si, 0.f);  // future_queue
            float inten = fgrow[j] + add - fq;
            pr[j] = fmaxf((1.0f - fmaxf(inten, 0.f) * 0.25f) - 0.6f, 0.f) + 0.6f;
        }
        if (active) {
            *(float4*)(o + i * 16 + 8)  = make_float4(pr[0], pr[1], pr[2], pr[3]);
            *(float4*)(o + i * 16 + 12) = make_float4(pr[4], pr[5], pr[6], pr[7]);
        }
    }
}

extern "C" void kernel_launch(void* const* d_in, const int* in_sizes, int n_in,
                              void* d_out, int out_size, void* d_ws, size_t ws_size,
                              hipStream_t stream) {
    const float* x  = (const float*)d_in[0];
    const float* W0 = (const float*)d_in[1];
    const float* b0 = (const float*)d_in[2];
    const float* W1 = (const float*)d_in[3];
    const float* b1 = (const float*)d_in[4];
    const float* dp = (const float*)d_in[5];
    const float* qp = (const float*)d_in[6];
    float* out = (float*)d_out;

    const int EP = in_sizes[0] / FF;
    const int blocks = (EP + BLOCK - 1) / BLOCK;
    actionnet_kernel<<<blocks, BLOCK, 0, stream>>>(x, W0, b0, W1, b1, dp, qp, out, EP);
}